// KernelWeightedMaeLoss_5188320493678
// MI455X (gfx1250) — compile-verified
//
#include <hip/hip_runtime.h>

// ---------------------------------------------------------------------------
// Weighted MAE loss with 3x3 neighborhood-min, CDNA5 (gfx1250).
// Memory-bound: ~402MB traffic -> ~17us at 23.3 TB/s. Strategy: stage pred
// tiles (with halo) into LDS via async global->LDS copies (ASYNCcnt path),
// stream target/mask with 128-bit loads, deterministic two-stage reduction.
// ---------------------------------------------------------------------------

#define HH   512
#define WW   512
#define SDIM 8
#define BDIM 16
#define TR   16           // rows per block tile
#define TPB  256          // 8 wave32s
#define W4   (WW / 4)     // 128 float4 per row
#define LROWS (TR + 2)    // tile rows incl. halo

typedef __attribute__((ext_vector_type(4))) int v4i;
typedef __attribute__((address_space(3))) v4i lds_v4i;

// ---- CDNA5 async global->LDS copy (tracked by ASYNCcnt) -------------------
__device__ __forceinline__ void async_copy_b128(const float* gp, float* lp) {
#if __has_builtin(__builtin_amdgcn_global_load_async_to_lds_b128)
  __builtin_amdgcn_global_load_async_to_lds_b128(
      (v4i*)gp,                 // global source (generic int4*, const stripped)
      (lds_v4i*)lp,             // LDS destination
      0, 0);                    // imm offset, cpol
#else
  unsigned loff = (unsigned)(unsigned long long)lp;   // LDS byte offset in low 32 bits
  asm volatile("global_load_async_to_lds_b128 %0, %1, off"
               :: "v"(loff), "v"(gp) : "memory");
#endif
}

__device__ __forceinline__ void wait_asynccnt0() {
#if __has_builtin(__builtin_amdgcn_s_wait_asynccnt)
  __builtin_amdgcn_s_wait_asynccnt(0);
#else
  asm volatile("s_wait_asynccnt 0" ::: "memory");
#endif
}

// ---------------------------------------------------------------------------
__global__ __launch_bounds__(TPB)
void wmae_main(const float* __restrict__ predicted,  // (S,B,H,W)
               const float* __restrict__ target,     // (B,S,H,W)
               const int*   __restrict__ mask,       // (B,S,H,W)
               float* __restrict__ partials) {
  __shared__ float tile[LROWS * WW];   // 36864 B
  __shared__ float wsum[TPB / 32];

  const int tid   = threadIdx.x;
  const int trow  = blockIdx.x;                 // 0..31 row tile
  const int plane = blockIdx.y;                 // 0..127 == b*S + s
  const int b = plane / SDIM;
  const int s = plane - b * SDIM;
  const int r0 = trow * TR;

  const unsigned long long tbase = (unsigned long long)plane * (HH * WW);
  const unsigned long long pbase = ((unsigned long long)s * BDIM + b) * (HH * WW);

  // ---- async-stage pred rows [r0-1, r0+TR] into LDS: 18*128 float4, 9/thread
  #pragma unroll
  for (int c = 0; c < (LROWS * W4) / TPB; ++c) {
    int v    = c * TPB + tid;
    int lrow = v >> 7;           // /128
    int c4   = v & 127;
    int gr   = r0 - 1 + lrow;
    if (gr >= 0 && gr < HH) {
      const float* gp = predicted + pbase + (unsigned long long)gr * WW + c4 * 4;
      async_copy_b128(gp, &tile[lrow * WW + c4 * 4]);
    }
  }
  wait_asynccnt0();
  __syncthreads();

  const float INF = __builtin_inff();
  float acc = 0.0f;

  // ---- 16 rows * 128 vec-cols = 2048 float4-pixels; 8 per thread ----------
  #pragma unroll
  for (int it = 0; it < (TR * W4) / TPB; ++it) {
    int v  = it * TPB + tid;
    int li = v >> 7;             // local row 0..15
    int c4 = v & 127;
    int i  = r0 + li;            // global row
    int j4 = c4 * 4;

    const float4 t4 = *(const float4*)(target + tbase + (unsigned long long)i * WW + j4);
    const int4   m4 = *(const int4*)(mask   + tbase + (unsigned long long)i * WW + j4);

    // gather 3 rows x [j4-1 .. j4+4] from LDS (tile row li..li+2 = rows i-1..i+1)
    float f[3][6];
    #pragma unroll
    for (int r = 0; r < 3; ++r) {
      const float* row = &tile[(li + r) * WW];
      float4 mid = *(const float4*)(row + j4);           // ds_load_b128
      f[r][0] = (j4 > 0)       ? row[j4 - 1] : 0.0f;
      f[r][1] = mid.x; f[r][2] = mid.y; f[r][3] = mid.z; f[r][4] = mid.w;
      f[r][5] = (j4 + 4 < WW)  ? row[j4 + 4] : 0.0f;
    }

    const bool rvTop = (i > 0), rvBot = (i < HH - 1);
    const float tv[4] = {t4.x, t4.y, t4.z, t4.w};
    const int   mv[4] = {m4.x, m4.y, m4.z, m4.w};

    #pragma unroll
    for (int k = 0; k < 4; ++k) {
      const float t = tv[k];
      float w = 1.0f;
      w += (t >= 2.0f)  ? 1.0f  : 0.0f;
      w += (t >= 5.0f)  ? 3.0f  : 0.0f;
      w += (t >= 10.0f) ? 5.0f  : 0.0f;
      w += (t >= 30.0f) ? 20.0f : 0.0f;
      w *= (float)mv[k];

      const bool lv = (k > 0) || (j4 > 0);          // col j-1 in range
      const bool rv = (k < 3) || (j4 + 4 < WW);     // col j+1 in range

      float best = INF;
      #pragma unroll
      for (int r = 0; r < 3; ++r) {
        float cl = lv ? fabsf(t - f[r][k])     : INF;
        float cc =      fabsf(t - f[r][k + 1]);
        float cr = rv ? fabsf(t - f[r][k + 2]) : INF;
        float rowmin = fminf(fminf(cl, cc), cr);
        bool rowok = (r == 0) ? rvTop : ((r == 2) ? rvBot : true);
        best = fminf(best, rowok ? rowmin : INF);
      }
      acc += w * best;   // w==0 -> 0; center always valid -> best finite
    }
  }

  // ---- deterministic block reduction (wave32 shuffles, fixed order) -------
  #pragma unroll
  for (int off = 16; off > 0; off >>= 1) acc += __shfl_down(acc, off, 32);
  const int wid = tid >> 5, lane = tid & 31;
  if (lane == 0) wsum[wid] = acc;
  __syncthreads();
  if (tid == 0) {
    float ssum = 0.0f;
    #pragma unroll
    for (int w = 0; w < TPB / 32; ++w) ssum += wsum[w];
    partials[blockIdx.y * gridDim.x + blockIdx.x] = ssum;
  }
}

// ---- final reduce: 4096 partials -> mean (f64 accumulate, deterministic) --
__global__ __launch_bounds__(256)
void wmae_reduce(const float* __restrict__ partials, int n, float* __restrict__ out) {
  __shared__ double sm[256];
  double acc = 0.0;
  for (int i = threadIdx.x; i < n; i += 256) acc += (double)partials[i];
  sm[threadIdx.x] = acc;
  __syncthreads();
  for (int s = 128; s > 0; s >>= 1) {
    if (threadIdx.x < s) sm[threadIdx.x] += sm[threadIdx.x + s];
    __syncthreads();
  }
  if (threadIdx.x == 0)
    out[0] = (float)(sm[0] / (double)((long long)BDIM * SDIM * HH * WW));
}

// ---------------------------------------------------------------------------
extern "C" void kernel_launch(void* const* d_in, const int* in_sizes, int n_in,
                              void* d_out, int out_size, void* d_ws, size_t ws_size,
                              hipStream_t stream) {
  const float* predicted = (const float*)d_in[0];   // (S,B,H,W) f32
  const float* target    = (const float*)d_in[1];   // (B,S,H,W) f32
  const int*   mask      = (const int*)d_in[2];     // (B,S,H,W) i32
  float* out      = (float*)d_out;
  float* partials = (float*)d_ws;                   // 4096 floats = 16 KB

  dim3 grid(HH / TR, BDIM * SDIM);                  // 32 x 128 = 4096 blocks
  wmae_main<<<grid, TPB, 0, stream>>>(predicted, target, mask, partials);

  const int nPart = (HH / TR) * BDIM * SDIM;
  wmae_reduce<<<1, 256, 0, stream>>>(partials, nPart, out);
}